// GATModel_627065225856
// MI455X (gfx1250) — compile-verified
//
#include <hip/hip_runtime.h>
#include <math.h>

// ---- problem constants (match reference) ----
#define NND   10000          // nodes
#define NED   160000         // edges (without self loops)
#define ETOT  (NED + NND)    // edges incl. self loops
#define NGR   64             // graphs
#define IND   128            // input dim
#define HIDC  64             // channels per head
#define NH    8              // heads (layers 0,1)
#define OUTD  10
#define D0    (NH * HIDC)    // 512

typedef float v2f __attribute__((ext_vector_type(2)));
typedef float v8f __attribute__((ext_vector_type(8)));

// ---- order-preserving float<->uint encoding for atomicMax-based segment max ----
__device__ __forceinline__ unsigned enc_f32(float x) {
  unsigned u = __float_as_uint(x);
  return (u & 0x80000000u) ? ~u : (u | 0x80000000u);
}
__device__ __forceinline__ float dec_f32(unsigned u) {
  unsigned b = (u & 0x80000000u) ? (u ^ 0x80000000u) : ~u;
  return __uint_as_float(b);
}
#define ENC_NEG_INF 0x007FFFFFu   // enc_f32(-inf)

// =====================================================================
// fp32 WMMA GEMM: C[M,N] = A[M,K] @ B[K,N]
// - A 16-row panel staged in LDS via GLOBAL_LOAD_ASYNC_TO_LDS_B128
//   (ASYNCcnt path), padded stride for bank spread.
// - one wave -> CT independent 16x16 accumulators (16 x 16*CT cols),
//   reusing each A fragment CT times; V_WMMA_F32_16X16X4_F32, K-step 4.
// - block = 128 threads (4 waves) covering 16 rows x 64*CT cols.
// =====================================================================
template <int KDIM, int NDIM, int CT>
__global__ __launch_bounds__(128) void gemm_wmma_f32(
    const float* __restrict__ A, const float* __restrict__ B,
    float* __restrict__ C)
{
  constexpr int LDA = KDIM + 4;              // pad (floats); keeps 16B alignment
  __shared__ float As[16 * LDA];

  const int row0 = blockIdx.x * 16;
  const int tid  = threadIdx.x;

  // ---- async-copy A panel (16 x KDIM floats) to LDS, 16B per lane ----
  constexpr int CHUNKS = 16 * KDIM / 4;      // multiple of 128 for our sizes
  for (int ci = tid; ci < CHUNKS; ci += 128) {
    const int r  = ci / (KDIM / 4);
    const int c4 = (ci % (KDIM / 4)) * 4;
    const unsigned lds = (unsigned)(size_t)(&As[r * LDA + c4]);  // addr[31:0] = LDS offset
    const float*   gp  = A + (size_t)(row0 + r) * KDIM + c4;
    asm volatile("global_load_async_to_lds_b128 %0, %1, off"
                 :: "v"(lds), "v"(gp) : "memory");
  }
  asm volatile("s_wait_asynccnt 0x0" ::: "memory");
  __syncthreads();

  const int wave = tid >> 5;
  const int lane = tid & 31;
  const int n    = lane & 15;          // A row / B,C col index for this lane
  const int kk   = (lane >> 4) * 2;    // K sub-offset per lane half
  const int colw = blockIdx.y * (64 * CT) + wave * (16 * CT);

  v8f acc[CT] = {};
  for (int k = 0; k < KDIM; k += 4) {
    v2f a;
    a.x = As[n * LDA + k + kk];
    a.y = As[n * LDA + k + kk + 1];
    #pragma unroll
    for (int t = 0; t < CT; ++t) {
      v2f b;
      b.x = B[(size_t)(k + kk)     * NDIM + colw + t * 16 + n];
      b.y = B[(size_t)(k + kk + 1) * NDIM + colw + t * 16 + n];
      acc[t] = __builtin_amdgcn_wmma_f32_16x16x4_f32(
          /*neg_a=*/false, a, /*neg_b=*/false, b,
          /*c_mod=*/(short)0, acc[t], /*reuse_a=*/false, /*reuse_b=*/false);
    }
  }

  const int mbase = (lane >> 4) * 8;   // C: lanes 0-15 -> rows 0..7, 16-31 -> 8..15
  #pragma unroll
  for (int t = 0; t < CT; ++t) {
    #pragma unroll
    for (int r = 0; r < 8; ++r)
      C[(size_t)(row0 + mbase + r) * NDIM + colw + t * 16 + n] = acc[t][r];
  }
}

// =====================================================================
// attention projections: a_s[n,h] = <h[n,h,:], asrc[h,:]>, same for a_d
// =====================================================================
__global__ void att_scores(const float* __restrict__ hw,
                           const float* __restrict__ asrc,
                           const float* __restrict__ adst,
                           float* __restrict__ a_s, float* __restrict__ a_d,
                           int H)
{
  const int i = blockIdx.x * blockDim.x + threadIdx.x;
  if (i >= NND * H) return;
  const int n = i / H, h = i % H;
  const float* hp = hw + (size_t)n * H * HIDC + h * HIDC;
  const float* as = asrc + h * HIDC;
  const float* ad = adst + h * HIDC;
  float s1 = 0.f, s2 = 0.f;
  #pragma unroll 4
  for (int c = 0; c < HIDC; ++c) { float v = hp[c]; s1 += v * as[c]; s2 += v * ad[c]; }
  a_s[i] = s1; a_d[i] = s2;
}

__device__ __forceinline__ void edge_sd(int e, const int* __restrict__ ei, int& s, int& d) {
  if (e < NED) { s = ei[e]; d = ei[NED + e]; }
  else         { s = d = e - NED; }          // appended self loops
}

// pass 1: e = leaky_relu(a_s[src]+a_d[dst]); segment max into menc[dst]
__global__ void edge_pass1(const int* __restrict__ ei,
                           const float* __restrict__ a_s, const float* __restrict__ a_d,
                           float* __restrict__ ebuf, unsigned* __restrict__ menc, int H)
{
  const int i = blockIdx.x * blockDim.x + threadIdx.x;
  if (i >= ETOT * H) return;
  const int e = i / H, h = i % H;
  int s, d; edge_sd(e, ei, s, d);
  float v = a_s[s * H + h] + a_d[d * H + h];
  v = v > 0.f ? v : 0.2f * v;
  ebuf[i] = v;
  atomicMax(&menc[d * H + h], enc_f32(v));
}

// pass 2: ex = exp(e - m[dst]); segment sum into ssum[dst]
__global__ void edge_pass2(const int* __restrict__ ei,
                           float* __restrict__ ebuf, const unsigned* __restrict__ menc,
                           float* __restrict__ ssum, int H)
{
  const int i = blockIdx.x * blockDim.x + threadIdx.x;
  if (i >= ETOT * H) return;
  const int e = i / H, h = i % H;
  int s, d; edge_sd(e, ei, s, d);
  const float ex = expf(ebuf[i] - dec_f32(menc[d * H + h]));
  ebuf[i] = ex;
  unsafeAtomicAdd(&ssum[d * H + h], ex);
}

// pass 3: out[dst] += h[src] * alpha  (out pre-initialized with bias)
__global__ void edge_pass3(const int* __restrict__ ei,
                           const float* __restrict__ ebuf, const float* __restrict__ ssum,
                           const float* __restrict__ hw, float* __restrict__ out, int H)
{
  const int i = blockIdx.x * blockDim.x + threadIdx.x;
  if (i >= ETOT * H) return;
  const int e = i / H, h = i % H;
  int s, d; edge_sd(e, ei, s, d);
  const float alpha = ebuf[i] / (ssum[d * H + h] + 1e-16f);
  const float* sp = hw + (size_t)s * H * HIDC + h * HIDC;
  float*       dp = out + (size_t)d * H * HIDC + h * HIDC;
  #pragma unroll 4
  for (int c = 0; c < HIDC; ++c) unsafeAtomicAdd(&dp[c], sp[c] * alpha);
}

// ---- init / fill / relu helpers ----
__global__ void fill_f32(float* p, float v, int n)      { int i = blockIdx.x*blockDim.x+threadIdx.x; if (i < n) p[i] = v; }
__global__ void fill_u32(unsigned* p, unsigned v, int n){ int i = blockIdx.x*blockDim.x+threadIdx.x; if (i < n) p[i] = v; }
__global__ void relu_inplace(float* p, int n)           { int i = blockIdx.x*blockDim.x+threadIdx.x; if (i < n) p[i] = fmaxf(p[i], 0.f); }
__global__ void init_bias(float* out, const float* __restrict__ b, int D) {
  int i = blockIdx.x*blockDim.x+threadIdx.x;
  if (i < NND * D) out[i] = b[i % D];
}

// ---- graph pooling: mean + max over nodes per graph ----
__global__ void pool_kernel(const float* __restrict__ h2, const int* __restrict__ batch,
                            float* __restrict__ sums, unsigned* __restrict__ mxenc,
                            float* __restrict__ cnt)
{
  const int i = blockIdx.x * blockDim.x + threadIdx.x;
  if (i >= NND * HIDC) return;
  const int n = i / HIDC, c = i % HIDC;
  const int g = batch[n];
  const float v = h2[i];
  unsafeAtomicAdd(&sums[g * HIDC + c], v);
  atomicMax(&mxenc[g * HIDC + c], enc_f32(v));
  if (c == 0) unsafeAtomicAdd(&cnt[g], 1.0f);
}

// ---- classifier: [mean|max](128) -> relu(@cW1+cb1)(64) -> @cW2+cb2(10) ----
__global__ __launch_bounds__(64) void classify(
    const float* __restrict__ sums, const unsigned* __restrict__ mxenc,
    const float* __restrict__ cnt,
    const float* __restrict__ cW1, const float* __restrict__ cb1,
    const float* __restrict__ cW2, const float* __restrict__ cb2,
    float* __restrict__ out)
{
  __shared__ float gv[2 * HIDC];
  __shared__ float hid[HIDC];
  const int g = blockIdx.x, t = threadIdx.x;
  const float c = fmaxf(cnt[g], 1.0f);
  gv[t]        = sums[g * HIDC + t] / c;
  gv[HIDC + t] = dec_f32(mxenc[g * HIDC + t]);
  __syncthreads();
  float acc = cb1[t];
  for (int i = 0; i < 2 * HIDC; ++i) acc += gv[i] * cW1[i * HIDC + t];
  hid[t] = acc > 0.f ? acc : 0.f;
  __syncthreads();
  if (t < OUTD) {
    float o = cb2[t];
    for (int i = 0; i < HIDC; ++i) o += hid[i] * cW2[i * OUTD + t];
    out[g * OUTD + t] = o;
  }
}

// =====================================================================
// driver
// =====================================================================
static inline int ceil_div(int a, int b) { return (a + b - 1) / b; }

extern "C" void kernel_launch(void* const* d_in, const int* in_sizes, int n_in,
                              void* d_out, int out_size, void* d_ws, size_t ws_size,
                              hipStream_t stream)
{
  const float* x     = (const float*)d_in[0];
  const int*   ei    = (const int*)  d_in[1];   // [2, NED]
  const int*   batch = (const int*)  d_in[2];
  const float* W0    = (const float*)d_in[3];
  const float* asrc0 = (const float*)d_in[4];
  const float* adst0 = (const float*)d_in[5];
  const float* b0    = (const float*)d_in[6];
  const float* W1    = (const float*)d_in[7];
  const float* asrc1 = (const float*)d_in[8];
  const float* adst1 = (const float*)d_in[9];
  const float* b1    = (const float*)d_in[10];
  const float* W2    = (const float*)d_in[11];
  const float* asrc2 = (const float*)d_in[12];
  const float* adst2 = (const float*)d_in[13];
  const float* b2    = (const float*)d_in[14];
  const float* cW1   = (const float*)d_in[15];
  const float* cb1   = (const float*)d_in[16];
  const float* cW2   = (const float*)d_in[17];
  const float* cb2   = (const float*)d_in[18];
  float* out = (float*)d_out;

  // workspace carve-up (floats)
  float*    hw    = (float*)d_ws;                       // [NND, 512]
  float*    agg   = hw   + (size_t)NND * D0;            // [NND, 512]
  float*    a_s   = agg  + (size_t)NND * D0;            // [NND, 8]
  float*    a_d   = a_s  + (size_t)NND * NH;
  float*    ebuf  = a_d  + (size_t)NND * NH;            // [ETOT, 8]
  float*    ssum  = ebuf + (size_t)ETOT * NH;           // [NND, 8]
  unsigned* menc  = (unsigned*)(ssum + (size_t)NND * NH);
  float*    sums  = (float*)(menc + (size_t)NND * NH);  // [64, 64]
  unsigned* mxenc = (unsigned*)(sums + NGR * HIDC);
  float*    cnt   = (float*)(mxenc + NGR * HIDC);       // [64]

  const int TB = 256;
  #define GRID(n) ceil_div((n), TB), TB, 0, stream

  // -------- GAT edge phase runner (host-side sequence) --------
  auto run_edges = [&](int H, const float* asrc, const float* adst,
                       const float* bias, float* feat, float* outbuf) {
    const int D = H * HIDC;
    att_scores<<<GRID(NND * H)>>>(feat, asrc, adst, a_s, a_d, H);
    fill_u32 <<<GRID(NND * H)>>>(menc, ENC_NEG_INF, NND * H);
    edge_pass1<<<GRID(ETOT * H)>>>(ei, a_s, a_d, ebuf, menc, H);
    fill_f32 <<<GRID(NND * H)>>>(ssum, 0.f, NND * H);
    edge_pass2<<<GRID(ETOT * H)>>>(ei, ebuf, menc, ssum, H);
    init_bias<<<GRID(NND * D)>>>(outbuf, bias, D);
    edge_pass3<<<GRID(ETOT * H)>>>(ei, ebuf, ssum, feat, outbuf, H);
  };

  // ---- layer 0: x[N,128] @ W0[128,512] ----
  gemm_wmma_f32<IND, D0, 4><<<dim3(NND / 16, D0 / 256), 128, 0, stream>>>(x, W0, hw);
  run_edges(NH, asrc0, adst0, b0, hw, agg);

  // ---- layer 1: relu(agg)[N,512] @ W1[512,512] ----
  relu_inplace<<<GRID(NND * D0)>>>(agg, NND * D0);
  gemm_wmma_f32<D0, D0, 4><<<dim3(NND / 16, D0 / 256), 128, 0, stream>>>(agg, W1, hw);
  run_edges(NH, asrc1, adst1, b1, hw, agg);

  // ---- layer 2: relu(agg)[N,512] @ W2[512,64], heads=1 ----
  relu_inplace<<<GRID(NND * D0)>>>(agg, NND * D0);
  gemm_wmma_f32<D0, HIDC, 1><<<dim3(NND / 16, 1), 128, 0, stream>>>(agg, W2, hw);
  run_edges(1, asrc2, adst2, b2, hw, agg);

  // ---- pooling + classifier ----
  fill_f32<<<GRID(NGR * HIDC)>>>(sums, 0.f, NGR * HIDC);
  fill_u32<<<GRID(NGR * HIDC)>>>(mxenc, ENC_NEG_INF, NGR * HIDC);
  fill_f32<<<GRID(NGR)>>>(cnt, 0.f, NGR);
  pool_kernel<<<GRID(NND * HIDC)>>>(agg, batch, sums, mxenc, cnt);
  classify<<<NGR, 64, 0, stream>>>(sums, mxenc, cnt, cW1, cb1, cW2, cb2, out);

  #undef GRID
}